// VisionTokenPruner_75634374082784
// MI455X (gfx1250) — compile-verified
//
#include <hip/hip_runtime.h>
#include <hip/hip_bf16.h>
#include <float.h>
#include <math.h>

// Problem constants (from reference): B=32, L=2048, D=1024, C=8
#define BB 32
#define LL 2048
#define DD 1024
#define CC 8
#define TT 614      // target_num = max(int(L*0.3), 15)
#define PV 153      // pivot_num  = T//4
#define PPAD 160    // pivots padded to multiple of 16 for WMMA N-tiles
#define KPICK 3     // topk_per_pivot
#define TAU_REL_C 0.1f
#define EPSN 1e-8f

// GEMM tiling
#define MBLK 128            // rows per block
#define KC   64             // K-chunk (floats)
#define NTILES (PPAD / 16)  // 10 N-tiles per block
#define A_ELEMS (MBLK * KC)             // 8192 floats = 32KB
#define B_ELEMS (PPAD * KC)             // 10240 floats = 40KB
#define BUF_ELEMS (A_ELEMS + B_ELEMS)   // 18432 floats = 72KB
#define SMEM_BYTES (2 * BUF_ELEMS * 4)  // 144KB double-buffered

typedef float v2f __attribute__((ext_vector_type(2)));
typedef float v8f __attribute__((ext_vector_type(8)));
typedef unsigned int u32x4 __attribute__((ext_vector_type(4)));
typedef int i32x4 __attribute__((ext_vector_type(4)));
typedef int i32x8 __attribute__((ext_vector_type(8)));

// ---------- helpers ----------
__device__ __forceinline__ unsigned orderF(float x) {
  unsigned u = __float_as_uint(x);
  return (u & 0x80000000u) ? ~u : (u | 0x80000000u);
}
// max-key packing: larger value wins; ties -> lower index wins (matches jax top_k)
__device__ __forceinline__ unsigned long long packKey(float v, int t) {
  return ((unsigned long long)orderF(v) << 32) | (unsigned)(~(unsigned)t);
}
__device__ __forceinline__ int unpackIdx(unsigned long long k) {
  return (int)(~(unsigned)(k & 0xFFFFFFFFull));
}
__device__ __forceinline__ int blockSum256(int v, int* sbuf) {
  int tid = threadIdx.x;
  sbuf[tid] = v; __syncthreads();
  for (int off = 128; off > 0; off >>= 1) {
    if (tid < off) sbuf[tid] += sbuf[tid + off];
    __syncthreads();
  }
  int r = sbuf[0]; __syncthreads();
  return r;
}
__device__ __forceinline__ int blockScanExcl256(int v, int* sbuf, int& total) {
  int tid = threadIdx.x;
  sbuf[tid] = v; __syncthreads();
  for (int off = 1; off < 256; off <<= 1) {
    int x = (tid >= off) ? sbuf[tid - off] : 0;
    __syncthreads();
    sbuf[tid] += x;
    __syncthreads();
  }
  int incl = sbuf[tid];
  total = sbuf[255];
  __syncthreads();
  return incl - v;
}

// ---------- TDM: issue a 2D-tile tensor_load_to_lds (D# per CDNA5 ISA ch.8) ----------
// data_size=4B, tile = tile_d0 x tile_d1 elements, row stride = stride0 elements.
__device__ __forceinline__ void tdm_load_2d(unsigned lds_off, unsigned long long gaddr,
                                            unsigned tensor_d0, unsigned tensor_d1,
                                            unsigned tile_d0, unsigned tile_d1,
                                            unsigned stride0) {
  u32x4 g0;
  g0.x = 1u;                                               // count=1, user mode, no gather
  g0.y = lds_off;                                          // lds_addr [63:32]
  g0.z = (unsigned)(gaddr & 0xFFFFFFFFull);                // global_addr[31:0]
  g0.w = ((unsigned)(gaddr >> 32) & 0x01FFFFFFu) | (2u << 30);  // addr[56:32] | type=2
  i32x8 g1;
  g1[0] = (int)(2u << 16);                                 // wg_mask=0, data_size=4B
  g1[1] = (int)((tensor_d0 & 0xFFFFu) << 16);              // tensor_dim0[15:0]
  g1[2] = (int)((tensor_d0 >> 16) | ((tensor_d1 & 0xFFFFu) << 16));
  g1[3] = (int)((tensor_d1 >> 16) | (tile_d0 << 16));      // tensor_dim1 hi | tile_dim0
  g1[4] = (int)tile_d1;                                    // tile_dim1 (tile_dim2=0)
  g1[5] = (int)stride0;                                    // tensor_dim0_stride[31:0]
  g1[6] = 0;
  g1[7] = 0;
  i32x4 z4 = {0, 0, 0, 0};
#if defined(__clang_major__) && (__clang_major__ >= 23)
  i32x8 z8 = {0, 0, 0, 0, 0, 0, 0, 0};
  __builtin_amdgcn_tensor_load_to_lds(g0, g1, z4, z4, z8, 0);
#else
  __builtin_amdgcn_tensor_load_to_lds(g0, g1, z4, z4, 0);
#endif
}

// ---------- K1: per-token rel (max posterior), norm, 1/clamped-norm, valid count ----------
__global__ void prep_kernel(const float* __restrict__ hidden,
                            const float* __restrict__ post,
                            float* __restrict__ rel, float* __restrict__ norms,
                            float* __restrict__ invn, int* __restrict__ cnt) {
  int wave = threadIdx.x >> 5, lane = threadIdx.x & 31;
  int tok = blockIdx.x * 8 + wave;          // 0 .. BB*LL-1
  int s = tok / LL;
  const float* row = hidden + (size_t)tok * DD;
  float ss = 0.f;
#pragma unroll
  for (int i = 0; i < DD / 32; ++i) { float x = row[i * 32 + lane]; ss += x * x; }
  for (int off = 16; off > 0; off >>= 1) ss += __shfl_xor(ss, off, 32);
  float r = (lane < CC) ? post[(size_t)tok * CC + lane] : -FLT_MAX;
  for (int off = 16; off > 0; off >>= 1) r = fmaxf(r, __shfl_xor(r, off, 32));
  if (lane == 0) {
    float nm = sqrtf(ss);
    norms[tok] = nm;
    invn[tok] = 1.f / fmaxf(nm, EPSN);
    rel[tok] = r;
    if (r > TAU_REL_C) atomicAdd(&cnt[s], 1);
  }
}

// ---------- K2: top-153 pivots by norm among valid tokens (descending, tie->low idx) ----------
__global__ void pivot_kernel(const float* __restrict__ rel,
                             const float* __restrict__ norms,
                             const float* __restrict__ invn,
                             int* __restrict__ pivots, float* __restrict__ invp) {
  int s = blockIdx.x, tid = threadIdx.x;
  __shared__ float key[LL];
  __shared__ unsigned long long red[256];
  for (int t = tid; t < LL; t += 256)
    key[t] = (rel[s * LL + t] > TAU_REL_C) ? norms[s * LL + t] : -1.f;
  __syncthreads();
  for (int it = 0; it < PV; ++it) {
    unsigned long long local = 0ull;
    for (int t = tid; t < LL; t += 256)
      if (key[t] >= 0.f) {
        unsigned long long pk = packKey(key[t], t);
        if (pk > local) local = pk;
      }
    red[tid] = local; __syncthreads();
    for (int off = 128; off > 0; off >>= 1) {
      if (tid < off) { if (red[tid + off] > red[tid]) red[tid] = red[tid + off]; }
      __syncthreads();
    }
    unsigned long long best = red[0];
    if (tid == 0) {
      if (best == 0ull) { pivots[s * PPAD + it] = -1; invp[s * PPAD + it] = 0.f; }
      else {
        int t = unpackIdx(best);
        pivots[s * PPAD + it] = t;
        invp[s * PPAD + it] = invn[s * LL + t];
        key[t] = -1.f;
      }
    }
    __syncthreads();
  }
  for (int it = PV + tid; it < PPAD; it += 256) {
    pivots[s * PPAD + it] = -1; invp[s * PPAD + it] = 0.f;
  }
}

// ---------- K3: gather pivot rows into compact [B,PPAD,D] (pad rows = 0) ----------
__global__ void gatherp_kernel(const float* __restrict__ hidden,
                               const int* __restrict__ pivots,
                               float* __restrict__ Hp) {
  int bid = blockIdx.x;                 // s*PPAD + j
  int s = bid / PPAD;
  int p = pivots[bid];
  float* dst = Hp + (size_t)bid * DD;
  const float* src = hidden + ((size_t)s * LL + (p < 0 ? 0 : p)) * DD;
  for (int i = threadIdx.x; i < DD; i += 256) dst[i] = (p >= 0) ? src[i] : 0.f;
}

// ---------- K4: LDS-tiled cosine GEMM: TDM double-buffered stage + V_WMMA_F32_16X16X4_F32 ----------
// Block = 8 waves, computes 128 rows x 160 pivot cols. Wave w: rows [w*16, w*16+16),
// 10 accumulators (one per N-tile) so each A fragment feeds 10 WMMAs.
// Per K-step: issue all 11 LDS fragment loads first, then the 10-WMMA chain, so the
// scheduler can stage s_wait_dscnt instead of round-tripping LDS latency per WMMA.
// A (16x4 f32): lanes 0-15 M=0..15 (K=0,1); lanes 16-31 same M, K=2,3.
// B (4x16 f32): lanes 0-15 N=0..15 (K=0,1); lanes 16-31 same N, K=2,3.
// D (16x16 f32): vgpr r, lanes 0-15 -> M=r, N=lane; lanes 16-31 -> M=r+8.
__global__ void __launch_bounds__(256) cosine_wmma_kernel(
    const float* __restrict__ hidden, const float* __restrict__ Hp,
    const float* __restrict__ invn, const float* __restrict__ invp,
    float* __restrict__ cosT) {
  extern __shared__ float smem[];
  const int NC = DD / KC;               // 16 K-chunks
  int mb = blockIdx.x, s = blockIdx.y;
  int tid = threadIdx.x, w = tid >> 5, lane = tid & 31, lo = lane & 15, hi = lane >> 4;
  bool issuer = (w == 0);

  unsigned lds_base = (unsigned)(uintptr_t)(void*)smem;   // low 32 bits = LDS offset
  unsigned ldsA[2] = { lds_base, lds_base + (unsigned)BUF_ELEMS * 4u };
  unsigned ldsB[2] = { ldsA[0] + (unsigned)A_ELEMS * 4u, ldsA[1] + (unsigned)A_ELEMS * 4u };

  unsigned long long gA =
      (unsigned long long)(uintptr_t)(hidden + ((size_t)s * LL + (size_t)mb * MBLK) * DD);
  unsigned long long gB = (unsigned long long)(uintptr_t)(Hp + (size_t)s * PPAD * DD);

  v8f acc[NTILES];
#pragma unroll
  for (int nt = 0; nt < NTILES; ++nt) acc[nt] = (v8f){0.f,0.f,0.f,0.f,0.f,0.f,0.f,0.f};

  if (issuer) {
    tdm_load_2d(ldsA[0], gA, DD, LL, KC, MBLK, DD);    // A chunk: 128 x 64
    tdm_load_2d(ldsB[0], gB, DD, PPAD, KC, PPAD, DD);  // B chunk: 160 x 64
  }
  for (int c = 0; c < NC; ++c) {
    int cur = c & 1;
    if (c + 1 < NC && issuer) {
      unsigned long long koff = (unsigned long long)(c + 1) * KC * 4ull;
      tdm_load_2d(ldsA[1 - cur], gA + koff, DD, LL, KC, MBLK, DD);
      tdm_load_2d(ldsB[1 - cur], gB + koff, DD, PPAD, KC, PPAD, DD);
    }
    if (issuer) {
      if (c + 1 < NC) __builtin_amdgcn_s_wait_tensorcnt(2);  // pair c landed, c+1 in flight
      else            __builtin_amdgcn_s_wait_tensorcnt(0);
    }
    __syncthreads();                    // buf[cur] visible to all waves

    const float* As = smem + (cur ? BUF_ELEMS : 0);
    const float* Bs = As + A_ELEMS;
    const float* ap = As + (w * 16 + lo) * KC + 2 * hi;
    const float* bp = Bs + lo * KC + 2 * hi;
#pragma unroll 2
    for (int kk = 0; kk < KC; kk += 4) {
      v2f a = *(const v2f*)(ap + kk);
      v2f b[NTILES];
#pragma unroll
      for (int nt = 0; nt < NTILES; ++nt)
        b[nt] = *(const v2f*)(bp + nt * 16 * KC + kk);
#pragma unroll
      for (int nt = 0; nt < NTILES; ++nt)
        acc[nt] = __builtin_amdgcn_wmma_f32_16x16x4_f32(false, a, false, b[nt],
                                                        (short)0, acc[nt], false, false);
    }
    __syncthreads();                    // done reading buf[cur] before it is re-filled
  }

  // epilogue: scale by 1/max(norm,eps) on both sides, store pivot-major
  int mrow = mb * MBLK + w * 16 + 8 * hi;
  const float* ia = invn + s * LL + mrow;
#pragma unroll
  for (int nt = 0; nt < NTILES; ++nt) {
    int n = nt * 16 + lo;
    float sb = invp[s * PPAD + n];
    float* outp = cosT + ((size_t)s * PPAD + n) * LL + mrow;
#pragma unroll
    for (int r2 = 0; r2 < 8; ++r2) outp[r2] = acc[nt][r2] * ia[r2] * sb;
  }
}

// ---------- K5: greedy suppression scan (branch B), one block per sample ----------
__global__ void greedy_kernel(const float* __restrict__ rel,
                              const int* __restrict__ cnt,
                              const int* __restrict__ pivots,
                              const float* __restrict__ cosT,
                              int* __restrict__ selIdx, int* __restrict__ selCnt) {
  int s = blockIdx.x, tid = threadIdx.x;
  if (cnt[s] <= TT) return;             // branch A sample
  __shared__ float col[LL];
  __shared__ unsigned char avail[LL], sel[LL];
  __shared__ unsigned long long red[256];
  __shared__ int sbuf[256];
  __shared__ int availCount, selCount;
  for (int t = tid; t < LL; t += 256) {
    avail[t] = (rel[s * LL + t] > TAU_REL_C) ? 1 : 0;
    sel[t] = 0;
  }
  __syncthreads();
  for (int j = tid; j < PV; j += 256) { // pivots are distinct & valid here
    int p = pivots[s * PPAD + j];
    sel[p] = 1; avail[p] = 0;
  }
  if (tid == 0) { availCount = cnt[s] - PV; selCount = PV; }
  __syncthreads();
  for (int j = 0; j < PV; ++j) {
    if (selCount >= TT || availCount <= 0) break;     // do-condition (monotone)
    const float* cp = cosT + ((size_t)s * PPAD + j) * LL;
    for (int t = tid; t < LL; t += 256) col[t] = cp[t];
    __syncthreads();
    for (int pick = 0; pick < KPICK; ++pick) {        // top-KP of (-cos) among avail
      if (availCount <= 0) break;
      unsigned long long local = 0ull;
      for (int t = tid; t < LL; t += 256)
        if (avail[t]) {
          unsigned long long pk = packKey(-col[t], t);
          if (pk > local) local = pk;
        }
      red[tid] = local; __syncthreads();
      for (int off = 128; off > 0; off >>= 1) {
        if (tid < off) { if (red[tid + off] > red[tid]) red[tid] = red[tid + off]; }
        __syncthreads();
      }
      unsigned long long best = red[0];
      if (tid == 0 && best != 0ull) {
        int bt = unpackIdx(best);
        sel[bt] = 1; avail[bt] = 0; selCount++; availCount--;
      }
      __syncthreads();
    }
  }
  // compact selected indices (ascending), clamp to T
  int base = tid * 8;
  int f[8]; int sum = 0;
#pragma unroll
  for (int i = 0; i < 8; ++i) { f[i] = sel[base + i]; sum += f[i]; }
  int total;
  int excl = blockScanExcl256(sum, sbuf, total);
  int run = excl;
#pragma unroll
  for (int i = 0; i < 8; ++i)
    if (f[i]) { if (run < TT) selIdx[s * TT + run] = base + i; run++; }
  int clamped = total < TT ? total : TT;
  for (int q = tid; q < TT; q += 256)
    if (q >= clamped) selIdx[s * TT + q] = 0;
  if (tid == 0) selCnt[s] = clamped;
}

// ---------- K6: branch A — top-T by rel, ties->low index, emitted in index order ----------
__global__ void brancha_kernel(const float* __restrict__ rel,
                               const int* __restrict__ cnt,
                               int* __restrict__ selIdx, int* __restrict__ selCnt) {
  int s = blockIdx.x, tid = threadIdx.x;
  if (cnt[s] > TT) return;
  __shared__ unsigned ok[LL];
  __shared__ int sbuf[256];
  for (int t = tid; t < LL; t += 256) ok[t] = orderF(rel[s * LL + t]);
  __syncthreads();
  // binary search for the T-th largest (largest v with countGE(v) >= T)
  unsigned v = 0u;
  for (int bit = 31; bit >= 0; --bit) {
    unsigned cand = v | (1u << bit);
    int c = 0;
    for (int t = tid; t < LL; t += 256) if (ok[t] >= cand) c++;
    int tot = blockSum256(c, sbuf);
    if (tot >= TT) v = cand;
  }
  int nG = 0;
  if (v != 0xFFFFFFFFu) {
    unsigned vp1 = v + 1u;
    int c = 0;
    for (int t = tid; t < LL; t += 256) if (ok[t] >= vp1) c++;
    nG = blockSum256(c, sbuf);
  }
  int quota = TT - nG;                  // how many ==v entries to take (lowest idx first)
  int base = tid * 8;
  int eq[8]; int esum = 0;
#pragma unroll
  for (int i = 0; i < 8; ++i) { eq[i] = (ok[base + i] == v); esum += eq[i]; }
  int etot;
  int eexcl = blockScanExcl256(esum, sbuf, etot);
  int inc[8]; int isum = 0; int erun = eexcl;
#pragma unroll
  for (int i = 0; i < 8; ++i) {
    int take = (ok[base + i] > v) || (eq[i] && erun < quota);
    if (eq[i]) erun++;
    inc[i] = take; isum += take;
  }
  int itot;
  int iexcl = blockScanExcl256(isum, sbuf, itot);
  int run = iexcl;
#pragma unroll
  for (int i = 0; i < 8; ++i)
    if (inc[i]) { if (run < TT) selIdx[s * TT + run] = base + i; run++; }
  for (int q = tid; q < TT; q += 256)
    if (q >= itot) selIdx[s * TT + q] = 0;      // safety fill
  if (tid == 0) selCnt[s] = TT;                 // branch A mask is all-true
}

// ---------- K7: write outputs ----------
__global__ void gather_out_kernel(const float* __restrict__ hidden,
                                  const int* __restrict__ selIdx,
                                  const int* __restrict__ selCnt,
                                  float* __restrict__ out) {
  int j = blockIdx.x, s = blockIdx.y;
  int ok = j < selCnt[s];
  int q = selIdx[s * TT + j];
  const float* src = hidden + ((size_t)s * LL + q) * DD;
  float* dst = out + ((size_t)s * TT + j) * DD;
  for (int i = threadIdx.x; i < DD; i += 256) dst[i] = ok ? src[i] : 0.f;
}

__global__ void idxmask_kernel(const int* __restrict__ selIdx,
                               const int* __restrict__ selCnt,
                               float* __restrict__ out) {
  int g = blockIdx.x * 256 + threadIdx.x;
  if (g >= BB * TT) return;
  int s = g / TT, j = g - s * TT;
  int ok = j < selCnt[s];
  out[(size_t)BB * TT * DD + g] = (float)selIdx[g];
  out[(size_t)BB * TT * DD + (size_t)BB * TT + g] = ok ? 1.f : 0.f;
}

// ---------- launch ----------
extern "C" void kernel_launch(void* const* d_in, const int* in_sizes, int n_in,
                              void* d_out, int out_size, void* d_ws, size_t ws_size,
                              hipStream_t stream) {
  (void)in_sizes; (void)n_in; (void)out_size; (void)ws_size;
  const float* hidden = (const float*)d_in[0];
  const float* post   = (const float*)d_in[1];
  float* out = (float*)d_out;

  char* w = (char*)d_ws;
  size_t off = 0;
  auto alloc = [&](size_t bytes) -> void* {
    void* p = w + off;
    off = (off + bytes + 255) & ~(size_t)255;
    return p;
  };
  float* rel    = (float*)alloc((size_t)BB * LL * 4);
  float* norms  = (float*)alloc((size_t)BB * LL * 4);
  float* invn   = (float*)alloc((size_t)BB * LL * 4);
  int*   cnt    = (int*)  alloc((size_t)BB * 4);
  int*   pivots = (int*)  alloc((size_t)BB * PPAD * 4);
  float* invp   = (float*)alloc((size_t)BB * PPAD * 4);
  int*   selIdx = (int*)  alloc((size_t)BB * TT * 4);
  int*   selCnt = (int*)  alloc((size_t)BB * 4);
  float* Hp     = (float*)alloc((size_t)BB * PPAD * DD * 4);   // 20 MB
  float* cosT   = (float*)alloc((size_t)BB * PPAD * LL * 4);   // 40 MB

  hipMemsetAsync(cnt, 0, BB * sizeof(int), stream);
  prep_kernel<<<BB * LL / 8, 256, 0, stream>>>(hidden, post, rel, norms, invn, cnt);
  pivot_kernel<<<BB, 256, 0, stream>>>(rel, norms, invn, pivots, invp);
  gatherp_kernel<<<BB * PPAD, 256, 0, stream>>>(hidden, pivots, Hp);
  cosine_wmma_kernel<<<dim3(LL / MBLK, BB), 256, SMEM_BYTES, stream>>>(hidden, Hp, invn, invp, cosT);
  greedy_kernel<<<BB, 256, 0, stream>>>(rel, cnt, pivots, cosT, selIdx, selCnt);
  brancha_kernel<<<BB, 256, 0, stream>>>(rel, cnt, selIdx, selCnt);
  gather_out_kernel<<<dim3(TT, BB), 256, 0, stream>>>(hidden, selIdx, selCnt, out);
  idxmask_kernel<<<(BB * TT + 255) / 256, 256, 0, stream>>>(selIdx, selCnt, out);
}